// SurfEval_89086211654048
// MI455X (gfx1250) — compile-verified
//
#include <hip/hip_runtime.h>
#include <hip/hip_bf16.h>

typedef __attribute__((ext_vector_type(2))) float v2f;
typedef __attribute__((ext_vector_type(8))) float v8f;

#define MM    128
#define NN    128
#define OUTU  1024
#define OUTV  1024
#define NB    4
#define UG    4      // u values per workgroup
#define WAVES 8      // 256 threads / wave32

// Fast reciprocal: v_rcp_f32 + one Newton-Raphson step (~0.5 ulp).
__device__ __forceinline__ float fast_rcp(float x) {
    float r = __builtin_amdgcn_rcpf(x);
    r = fmaf(fmaf(-x, r, 1.0f), r, r);
    return r;
}

__global__ __launch_bounds__(256) void surf_eval_wmma(
    const float* __restrict__ ctrl,   // [NB][128][128][4]
    const float* __restrict__ Nu,     // [1024][4]
    const float* __restrict__ Nv,     // [1024][4]
    const int*   __restrict__ uspan,  // [1024]
    const int*   __restrict__ vspan,  // [1024]
    float*       __restrict__ out)    // [NB][1024][1024][3]
{
    // LDS: u-contracted slab tu[ul][n][d] (4*128*4 f32 = 8KB) + per-wave D tiles (8KB)
    __shared__ float4 tu_s4[UG * NN];          // float4 over d
    __shared__ float  d_s[WAVES][16 * 16];

    const float* tu_s = (const float*)tu_s4;

    const int tid = threadIdx.x;
    const int wg  = blockIdx.x;
    const int b   = wg >> 8;        // 0..3
    const int ug  = wg & 255;       // 0..255
    const int u0  = ug * UG;

    // ---------------- Stage 1: tu[ul][n][:] = sum_l Nu[u0+ul][l] * ctrl[b][uspan-3+l][n][:]
    const float4* ctrl4 = (const float4*)ctrl;
    const float4* Nu4   = (const float4*)Nu;
    #pragma unroll
    for (int it = 0; it < 2; ++it) {
        int e  = tid + 256 * it;    // 0..511 -> (ul, n)
        int ul = e >> 7;            // 0..3
        int n  = e & 127;
        int u  = u0 + ul;
        int sp = uspan[u];          // in [3,127]
        float4 w = Nu4[u];
        float wl[4] = {w.x, w.y, w.z, w.w};
        float ax = 0.f, ay = 0.f, az = 0.f, aw = 0.f;
        #pragma unroll
        for (int l = 0; l < 4; ++l) {
            float4 cp = ctrl4[(b * MM + (sp - 3 + l)) * NN + n];
            ax = fmaf(wl[l], cp.x, ax);
            ay = fmaf(wl[l], cp.y, ay);
            az = fmaf(wl[l], cp.z, az);
            aw = fmaf(wl[l], cp.w, aw);
        }
        float4 r; r.x = ax; r.y = ay; r.z = az; r.w = aw;
        tu_s4[e] = r;
    }
    __syncthreads();

    // ---------------- Stage 2: per 16-v tile, D(16x16) = A(16x16) x B(16x16) via 4x WMMA K=4
    const int wave = tid >> 5;
    const int lane = tid & 31;
    const int vloc = lane & 15;     // A row (v), and B/C/D column index
    const int half = lane >> 4;     // 0 or 1
    const int ulc  = vloc >> 2;     // B column -> (u_local, d)
    const int dc   = vloc & 3;

    const float4* Nv4 = (const float4*)Nv;

    for (int it = 0; it < 8; ++it) {
        int vt = wave + WAVES * it;     // 0..63
        int v0 = vt * 16;
        int v  = v0 + vloc;

        // K window base over n; spans are nondecreasing so vspan[v0] is the tile min.
        int nmin = vspan[v0] - 3;
        if (nmin > NN - 16) nmin = NN - 16;   // keep window inside [0,127]
        int off = vspan[v] - 3 - nmin;        // first nonzero k for this row, 0..12

        float4 nv = Nv4[v];

        v8f c = {};   // f32 accumulator, D = sum_s A_s * B_s
        #pragma unroll
        for (int s = 0; s < 4; ++s) {
            // A-matrix 16x4 f32 layout: lane<16 holds K={4s+0,4s+1}, lane>=16 K={4s+2,4s+3}
            int k0 = 4 * s + half * 2;
            int k1 = k0 + 1;
            int r0 = k0 - off;
            int r1 = k1 - off;
            float a0 = (r0 == 0) ? nv.x : (r0 == 1) ? nv.y : (r0 == 2) ? nv.z : (r0 == 3) ? nv.w : 0.0f;
            float a1 = (r1 == 0) ? nv.x : (r1 == 1) ? nv.y : (r1 == 2) ? nv.z : (r1 == 3) ? nv.w : 0.0f;
            v2f A; A[0] = a0; A[1] = a1;

            // B-matrix 4x16 f32: row k striped across lanes (col = lane%16),
            // VGPR0 rows {k0}, VGPR1 rows {k1} per half (mirrors A's K split).
            int n0 = nmin + k0;
            v2f B;
            B[0] = tu_s[(ulc * NN + n0) * 4 + dc];
            B[1] = tu_s[(ulc * NN + n0 + 1) * 4 + dc];

            c = __builtin_amdgcn_wmma_f32_16x16x4_f32(
                    /*neg_a=*/false, A, /*neg_b=*/false, B,
                    /*c_mod=*/(short)0, c, /*reuse_a=*/false, /*reuse_b=*/false);
        }

        // D layout: VGPR j holds row M = j + half*8, col = lane%16.
        // Stage through per-wave LDS to get the perspective divide + coalesced stores.
        #pragma unroll
        for (int j = 0; j < 8; ++j)
            d_s[wave][(j + half * 8) * 16 + vloc] = c[j];

        asm volatile("s_wait_dscnt 0" ::: "memory");  // intra-wave LDS write->read

        // 192 outputs per tile: 4 u * 16 v * 3 dims, contiguous 48-float chunk per u.
        #pragma unroll
        for (int t = 0; t < 6; ++t) {
            int e   = lane + 32 * t;   // 0..191
            int ulw = e / 48;
            int rem = e - 48 * ulw;
            int vw  = rem / 3;
            int dw  = rem - 3 * vw;
            float num = d_s[wave][vw * 16 + ulw * 4 + dw];
            float den = d_s[wave][vw * 16 + ulw * 4 + 3];
            size_t oi = ((((size_t)b * OUTU) + (u0 + ulw)) * OUTV + (v0 + vw)) * 3 + dw;
            out[oi] = num * fast_rcp(den);
        }
        // Next iteration's d_s writes are LDS-in-order after these reads; no extra wait.
    }
}

extern "C" void kernel_launch(void* const* d_in, const int* in_sizes, int n_in,
                              void* d_out, int out_size, void* d_ws, size_t ws_size,
                              hipStream_t stream) {
    const float* ctrl  = (const float*)d_in[0];
    const float* Nu    = (const float*)d_in[1];
    const float* Nv    = (const float*)d_in[2];
    const int*   uspan = (const int*)d_in[3];
    const int*   vspan = (const int*)d_in[4];
    float*       out   = (float*)d_out;

    surf_eval_wmma<<<dim3(NB * (OUTU / UG)), dim3(256), 0, stream>>>(
        ctrl, Nu, Nv, uspan, vspan, out);
}